// AdaGMNConv_25323127177620
// MI455X (gfx1250) — compile-verified
//
#include <hip/hip_runtime.h>
#include <hip/hip_bf16.h>
#include <math.h>

typedef float v2f __attribute__((ext_vector_type(2)));
typedef float v8f __attribute__((ext_vector_type(8)));

#define D       128
#define G_      8
#define NPER    4096
#define MPER    2048
#define QCHUNKS 128          // 2048 / 16
#define QSTR    132          // LDS row stride for 16x128 tiles (pad, 16B-aligned rows)
#define SSTR    18           // LDS row stride for 16x16 scale tile (8B-aligned)

__device__ __forceinline__ v8f wmma_f32(v2f a, v2f b, v8f c) {
    // V_WMMA_F32_16X16X4_F32 : D = A(16x4) * B(4x16) + C(16x16), fp32 exact
    return __builtin_amdgcn_wmma_f32_16x16x4_f32(false, a, false, b, (short)0, c,
                                                 false, false);
}

// ---------------------------------------------------------------------------
// Kernel 0a: transpose W1 (256x128) -> W1T (128x256), W2 (128x128) -> W2T
// so MLP B-fragments (consecutive k, fixed d) become single b64 loads.
// ---------------------------------------------------------------------------
__global__ __launch_bounds__(256)
void wtrans_kernel(const float* __restrict__ W1, const float* __restrict__ W2,
                   float* __restrict__ W1T, float* __restrict__ W2T) {
    const int d = blockIdx.x;        // 0..127 output row of transposed
    const int k = threadIdx.x;       // 0..255
    W1T[(size_t)d * 256 + k] = W1[(size_t)k * D + d];
    if (k < 128) W2T[(size_t)d * 128 + k] = W2[(size_t)k * D + d];
}

// ---------------------------------------------------------------------------
// Kernel 0b: tail sums over low-degree rows -> initialize p1 / p2
//   p2[g] = sum x[g*4096+2048 : g*4096+4096]
//   p1[g] = sum x[2048 : 4096]              (graph 0 tail, same for all g)
// ---------------------------------------------------------------------------
__global__ __launch_bounds__(128)
void tails_kernel(const float* __restrict__ x, float* __restrict__ p1,
                  float* __restrict__ p2) {
    const int g = blockIdx.x;
    const int d = threadIdx.x;
    float s_own = 0.f, s_g0 = 0.f;
    for (int r = 0; r < MPER; ++r) {
        s_own += x[((size_t)g * NPER + MPER + r) * D + d];
        s_g0  += x[((size_t)MPER + r) * D + d];
    }
    p2[g * D + d] = s_own;
    p1[g * D + d] = s_g0;
}

// ---------------------------------------------------------------------------
// Kernel 1: streaming softmax stats over a = P @ Q^T  (per P-row max / sumexp)
//   mode 0 (blocks 0..1023):    P = hdx 16-row block, Q = gxf    -> rowmax/rowsum[m]
//   mode 1 (blocks 1024..2047): P = gxf 16-row block, Q = hdx(g) -> colmax/colsum[g,j]
// One wave per block; 32x f32 WMMA per 16x16 tile, online (max,sum) merge.
// ---------------------------------------------------------------------------
__global__ __launch_bounds__(32)
void attn_stats_kernel(const float* __restrict__ x,
                       float* __restrict__ rowMax, float* __restrict__ rowSum,
                       float* __restrict__ colMax, float* __restrict__ colSum) {
    __shared__ float tileQ[16 * QSTR];
    const int lane = threadIdx.x;
    const int r = lane & 15, hi = lane >> 4;
    const int mode = blockIdx.x >> 10;
    const int b = blockIdx.x & 1023;
    const int g = b >> 7, t = b & 127;
    const float* P = mode ? (x + (size_t)(t * 16) * D)
                          : (x + ((size_t)g * NPER + t * 16) * D);
    const float* Q = mode ? (x + (size_t)g * NPER * D) : x;
    float* outMax = mode ? colMax : rowMax;
    float* outSum = mode ? colSum : rowSum;

    // A fragments: P rows striped per ISA 16x4 layout (loaded once, reused 128x)
    v2f afrag[32];
#pragma unroll
    for (int s = 0; s < 32; ++s)
        afrag[s] = *reinterpret_cast<const v2f*>(P + (size_t)r * D + 4 * s + 2 * hi);

    float rmax[8], rsum[8];
#pragma unroll
    for (int v = 0; v < 8; ++v) { rmax[v] = -3.402823466e38f; rsum[v] = 0.f; }

    for (int qc = 0; qc < QCHUNKS; ++qc) {
        const float* qsrc = Q + (size_t)qc * 16 * D;
#pragma unroll
        for (int i = 0; i < 16; ++i) {            // stage 16x128 chunk to LDS
            int f = lane + 32 * i;                // float4 index 0..511
            int row = f >> 5, c4 = f & 31;
            float4 val = *reinterpret_cast<const float4*>(qsrc + (size_t)row * D + c4 * 4);
            *reinterpret_cast<float4*>(&tileQ[row * QSTR + c4 * 4]) = val;
        }
        __syncthreads();
        if (qc + 1 < QCHUNKS) __builtin_prefetch(qsrc + 16 * D + lane * 16, 0, 1);

        v8f c = {};
#pragma unroll
        for (int sg = 0; sg < 4; ++sg) {          // a-tile = P @ Q^T (K=128)
            v2f bf[8];                            // batch B-frags, then 8 WMMAs
#pragma unroll
            for (int u = 0; u < 8; ++u)
                bf[u] = *reinterpret_cast<const v2f*>(
                    &tileQ[r * QSTR + 4 * (sg * 8 + u) + 2 * hi]);
#pragma unroll
            for (int u = 0; u < 8; ++u)
                c = wmma_f32(afrag[sg * 8 + u], bf[u], c);
        }
        // per-row online (max, sumexp): reduce across the 16 lanes of each half
#pragma unroll
        for (int v = 0; v < 8; ++v) {
            float e = c[v];
            float m = e;
            for (int off = 8; off >= 1; off >>= 1) m = fmaxf(m, __shfl_xor(m, off, 32));
            float nm = fmaxf(rmax[v], m);
            float se = __expf(e - nm);
            for (int off = 8; off >= 1; off >>= 1) se += __shfl_xor(se, off, 32);
            rsum[v] = rsum[v] * __expf(rmax[v] - nm) + se;
            rmax[v] = nm;
        }
        __syncthreads();
    }
    if (r == 0) {
#pragma unroll
        for (int v = 0; v < 8; ++v) {
            int row = v + 8 * hi;                 // rows 0..7 (lane0), 8..15 (lane16)
            outMax[(size_t)b * 16 + row] = rmax[v];
            outSum[(size_t)b * 16 + row] = rsum[v];
        }
    }
}

// ---------------------------------------------------------------------------
// Kernel 2: fused  attn-weighted sum -> MLP -> per-graph add-pool (atomics)
//   mode 0: P=hdx block, Q=gxf, scale by colmax/colsum (+1e-16) -> pool p2[g]
//   mode 1: P=gxf block, Q=hdx(g), scale by rowmax/rowsum       -> pool p1[g]
// ---------------------------------------------------------------------------
__global__ __launch_bounds__(32)
void attn_mlp_pool_kernel(const float* __restrict__ x, const float* __restrict__ W1T,
                          const float* __restrict__ b1, const float* __restrict__ gamma,
                          const float* __restrict__ beta, const float* __restrict__ W2T,
                          const float* __restrict__ b2,
                          const float* __restrict__ rowMax, const float* __restrict__ rowSum,
                          const float* __restrict__ colMax, const float* __restrict__ colSum,
                          float* __restrict__ p1, float* __restrict__ p2) {
    __shared__ float tileQ[16 * QSTR];
    __shared__ float tileS[16 * SSTR];
    __shared__ float tileT[16 * QSTR];            // mu tile, then h tile
    const int lane = threadIdx.x;
    const int r = lane & 15, hi = lane >> 4;
    const int mode = blockIdx.x >> 10;
    const int b = blockIdx.x & 1023;
    const int g = b >> 7, t = b & 127;
    const float* P = mode ? (x + (size_t)(t * 16) * D)
                          : (x + ((size_t)g * NPER + t * 16) * D);
    const float* Q = mode ? (x + (size_t)g * NPER * D) : x;
    const float* statMax = mode ? rowMax : colMax;
    const float* statSum = mode ? rowSum : colSum;
    const float denomEps = mode ? 0.f : 1e-16f;
    float* poolg = (mode ? p1 : p2) + g * D;
    const int statBase = g * MPER;

    v2f afrag[32];                                // P rows: A frags + MLP first half
#pragma unroll
    for (int s = 0; s < 32; ++s)
        afrag[s] = *reinterpret_cast<const v2f*>(P + (size_t)r * D + 4 * s + 2 * hi);

    v8f acc[8];                                   // mu tile [16 x 128], 8 N-tiles
#pragma unroll
    for (int nt = 0; nt < 8; ++nt) acc[nt] = (v8f){};

    for (int qc = 0; qc < QCHUNKS; ++qc) {
        const float* qsrc = Q + (size_t)qc * 16 * D;
#pragma unroll
        for (int i = 0; i < 16; ++i) {
            int f = lane + 32 * i;
            int row = f >> 5, c4 = f & 31;
            float4 val = *reinterpret_cast<const float4*>(qsrc + (size_t)row * D + c4 * 4);
            *reinterpret_cast<float4*>(&tileQ[row * QSTR + c4 * 4]) = val;
        }
        __syncthreads();
        if (qc + 1 < QCHUNKS) __builtin_prefetch(qsrc + 16 * D + lane * 16, 0, 1);

        v8f c = {};
#pragma unroll
        for (int sg = 0; sg < 4; ++sg) {          // recompute a-tile (K=128)
            v2f bf[8];
#pragma unroll
            for (int u = 0; u < 8; ++u)
                bf[u] = *reinterpret_cast<const v2f*>(
                    &tileQ[r * QSTR + 4 * (sg * 8 + u) + 2 * hi]);
#pragma unroll
            for (int u = 0; u < 8; ++u)
                c = wmma_f32(afrag[sg * 8 + u], bf[u], c);
        }
        // scale to attention weights (stats indexed by streamed column)
        const int ci = statBase + qc * 16 + r;
        const float sm = statMax[ci];
        const float sd = statSum[ci] + denomEps;
#pragma unroll
        for (int v = 0; v < 8; ++v)               // transpose S into A-layout LDS
            tileS[(v + 8 * hi) * SSTR + r] = __expf(c[v] - sm) / sd;
        __syncthreads();
        // acc += S(16x16) @ Qchunk(16x128)
#pragma unroll
        for (int s4 = 0; s4 < 4; ++s4) {
            v2f aS = *reinterpret_cast<const v2f*>(&tileS[r * SSTR + 4 * s4 + 2 * hi]);
            v2f bS[8];
#pragma unroll
            for (int nt = 0; nt < 8; ++nt) {
                bS[nt].x = tileQ[(4 * s4 + 2 * hi) * QSTR + nt * 16 + r];
                bS[nt].y = tileQ[(4 * s4 + 2 * hi + 1) * QSTR + nt * 16 + r];
            }
#pragma unroll
            for (int nt = 0; nt < 8; ++nt)
                acc[nt] = wmma_f32(aS, bS[nt], acc[nt]);
        }
        __syncthreads();
    }

    // ---- MLP: h = [P | mu] @ W1 + b1 ; LayerNorm ; ReLU ; @ W2 + b2 ----
#pragma unroll
    for (int nt = 0; nt < 8; ++nt)                // mu tile (C layout) -> LDS
#pragma unroll
        for (int v = 0; v < 8; ++v)
            tileT[(v + 8 * hi) * QSTR + nt * 16 + r] = acc[nt][v];
    __syncthreads();

    v8f h[8];
#pragma unroll
    for (int nt = 0; nt < 8; ++nt) h[nt] = (v8f){};
#pragma unroll
    for (int s = 0; s < 64; ++s) {                // K = 256
        v2f aH = (s < 32)
                     ? afrag[s]
                     : *reinterpret_cast<const v2f*>(&tileT[r * QSTR + 4 * (s - 32) + 2 * hi]);
        const int k0 = 4 * s + 2 * hi;
        v2f bW[8];
#pragma unroll
        for (int nt = 0; nt < 8; ++nt)            // W1T[d][k]: contiguous k pair
            bW[nt] = *reinterpret_cast<const v2f*>(W1T + (size_t)(nt * 16 + r) * 256 + k0);
#pragma unroll
        for (int nt = 0; nt < 8; ++nt)
            h[nt] = wmma_f32(aH, bW[nt], h[nt]);
    }
    float bb1[8], gam[8], bet[8], bb2[8];
#pragma unroll
    for (int nt = 0; nt < 8; ++nt) {
        bb1[nt] = b1[nt * 16 + r];  gam[nt] = gamma[nt * 16 + r];
        bet[nt] = beta[nt * 16 + r]; bb2[nt] = b2[nt * 16 + r];
    }
#pragma unroll
    for (int nt = 0; nt < 8; ++nt)
#pragma unroll
        for (int v = 0; v < 8; ++v) h[nt][v] += bb1[nt];
    // LayerNorm per row (row v / v+8 lives across 16 lanes x 8 N-tiles)
#pragma unroll
    for (int v = 0; v < 8; ++v) {
        float s = 0.f;
#pragma unroll
        for (int nt = 0; nt < 8; ++nt) s += h[nt][v];
        for (int off = 8; off >= 1; off >>= 1) s += __shfl_xor(s, off, 32);
        const float mu_ = s * (1.f / 128.f);
        float q = 0.f;
#pragma unroll
        for (int nt = 0; nt < 8; ++nt) { float dd = h[nt][v] - mu_; q += dd * dd; }
        for (int off = 8; off >= 1; off >>= 1) q += __shfl_xor(q, off, 32);
        const float inv = rsqrtf(q * (1.f / 128.f) + 1e-5f);
#pragma unroll
        for (int nt = 0; nt < 8; ++nt)
            h[nt][v] = fmaxf((h[nt][v] - mu_) * inv * gam[nt] + bet[nt], 0.f);
    }
    __syncthreads();
#pragma unroll
    for (int nt = 0; nt < 8; ++nt)                // h tile -> LDS for A-layout
#pragma unroll
        for (int v = 0; v < 8; ++v)
            tileT[(v + 8 * hi) * QSTR + nt * 16 + r] = h[nt][v];
    __syncthreads();

    v8f o[8];
#pragma unroll
    for (int nt = 0; nt < 8; ++nt) o[nt] = (v8f){};
#pragma unroll
    for (int s = 0; s < 32; ++s) {                // @ W2, K = 128
        v2f aH = *reinterpret_cast<const v2f*>(&tileT[r * QSTR + 4 * s + 2 * hi]);
        const int k0 = 4 * s + 2 * hi;
        v2f bW[8];
#pragma unroll
        for (int nt = 0; nt < 8; ++nt)            // W2T[d][k]: contiguous k pair
            bW[nt] = *reinterpret_cast<const v2f*>(W2T + (size_t)(nt * 16 + r) * 128 + k0);
#pragma unroll
        for (int nt = 0; nt < 8; ++nt)
            o[nt] = wmma_f32(aH, bW[nt], o[nt]);
    }
    // column add-pool (16 rows) -> global atomics into p1/p2
#pragma unroll
    for (int nt = 0; nt < 8; ++nt) {
        float cs = 0.f;
#pragma unroll
        for (int v = 0; v < 8; ++v) cs += o[nt][v] + bb2[nt];
        cs += __shfl_xor(cs, 16, 32);             // combine row halves
        if (lane < 16) atomicAdd(&poolg[nt * 16 + lane], cs);
    }
}

// ---------------------------------------------------------------------------
// Kernel 3: out[g] = (p1.p2) / (max(|p1|,eps) * max(|p2|,eps))
// ---------------------------------------------------------------------------
__global__ __launch_bounds__(128)
void final_kernel(const float* __restrict__ p1, const float* __restrict__ p2,
                  float* __restrict__ out) {
    __shared__ float s11[128], s22[128], s12[128];
    const int d = threadIdx.x;
    for (int g = 0; g < G_; ++g) {
        float a = p1[g * D + d], b = p2[g * D + d];
        s11[d] = a * a; s22[d] = b * b; s12[d] = a * b;
        __syncthreads();
        for (int off = 64; off >= 1; off >>= 1) {
            if (d < off) {
                s11[d] += s11[d + off]; s22[d] += s22[d + off]; s12[d] += s12[d + off];
            }
            __syncthreads();
        }
        if (d == 0) {
            float n1 = fmaxf(sqrtf(s11[0]), 1e-12f);
            float n2 = fmaxf(sqrtf(s22[0]), 1e-12f);
            out[g] = s12[0] / (n1 * n2);
        }
        __syncthreads();
    }
}

extern "C" void kernel_launch(void* const* d_in, const int* in_sizes, int n_in,
                              void* d_out, int out_size, void* d_ws, size_t ws_size,
                              hipStream_t stream) {
    (void)in_sizes; (void)n_in; (void)out_size; (void)ws_size;
    const float* x     = (const float*)d_in[0];
    const float* W1    = (const float*)d_in[2];
    const float* b1    = (const float*)d_in[3];
    const float* gamma = (const float*)d_in[4];
    const float* beta  = (const float*)d_in[5];
    const float* W2    = (const float*)d_in[6];
    const float* b2    = (const float*)d_in[7];
    float* out = (float*)d_out;

    float* ws     = (float*)d_ws;                 // ~460 KB total
    float* rowmax = ws;                           // 16384
    float* rowsum = ws + 16384;                   // 16384
    float* colmax = ws + 32768;                   // 16384  [g][j]
    float* colsum = ws + 49152;                   // 16384
    float* p1     = ws + 65536;                   // 8*128
    float* p2     = ws + 66560;                   // 8*128
    float* w1t    = ws + 67584;                   // 128*256
    float* w2t    = ws + 100352;                  // 128*128

    wtrans_kernel<<<128, 256, 0, stream>>>(W1, W2, w1t, w2t);
    tails_kernel<<<G_, 128, 0, stream>>>(x, p1, p2);
    attn_stats_kernel<<<2048, 32, 0, stream>>>(x, rowmax, rowsum, colmax, colsum);
    attn_mlp_pool_kernel<<<2048, 32, 0, stream>>>(x, w1t, b1, gamma, beta, w2t, b2,
                                                  rowmax, rowsum, colmax, colsum, p1, p2);
    final_kernel<<<1, 128, 0, stream>>>(p1, p2, out);
}